// TwoWordDepdProbe_44744969290267
// MI455X (gfx1250) — compile-verified
//
#include <hip/hip_runtime.h>
#include <hip/hip_bf16.h>

// Problem constants (match reference): B=16, L=256, D=1024, R=128
#define BB 16
#define LL 256
#define DD 1024
#define RR 128
#define ROWS (BB * LL)        // 4096 rows of t
#define OUT_STRIDE (LL * LL)  // 65536 per batch

typedef float v2f __attribute__((ext_vector_type(2)));
typedef float v8f __attribute__((ext_vector_type(8)));

// ---------------------------------------------------------------------------
// Kernel 1: t = batch @ proj   ([4096,1024] x [1024,128] -> [4096,128] f32)
// One wave computes a 16x32 strip of t (two 16x16 N-tiles) so the A fragment
// is reused by 2 WMMAs per K-step: 2 wmma per 5 vmem issues.
// 1024 waves -> 128 blocks of 256 threads.
// ---------------------------------------------------------------------------
__global__ __launch_bounds__(256) void proj_wmma_kernel(
    const float* __restrict__ batch,  // [4096, 1024]
    const float* __restrict__ proj,   // [1024, 128]
    float* __restrict__ t)            // [4096, 128]
{
  const int lane = threadIdx.x & 31;
  const int wid  = blockIdx.x * 8 + (threadIdx.x >> 5);  // 0..1023
  const int tileM = wid >> 2;   // 0..255  (16-row strips of t)
  const int tnp   = wid & 3;    // 0..3    (32-col strips of t)

  const int m    = lane & 15;        // row (A) / col (B) within tile
  const int koff = (lane >> 4) * 2;  // K pair: 0 or 2

  const float* arow  = batch + (size_t)(tileM * 16 + m) * DD;  // A row, contiguous K
  const float* bcol0 = proj + (tnp * 32 + m);       // B col for N-tile 2*tnp
  const float* bcol1 = proj + (tnp * 32 + 16 + m);  // B col for N-tile 2*tnp+1

  v8f acc0 = {0.f, 0.f, 0.f, 0.f, 0.f, 0.f, 0.f, 0.f};
  v8f acc1 = {0.f, 0.f, 0.f, 0.f, 0.f, 0.f, 0.f, 0.f};

  #pragma unroll 4
  for (int k = 0; k < DD; k += 4) {
    v2f a = *(const v2f*)(arow + k + koff);   // A[m][k+koff .. +1]
    v2f b0, b1;
    b0.x = bcol0[(size_t)(k + koff) * RR];
    b0.y = bcol0[(size_t)(k + koff + 1) * RR];
    b1.x = bcol1[(size_t)(k + koff) * RR];
    b1.y = bcol1[(size_t)(k + koff + 1) * RR];
    acc0 = __builtin_amdgcn_wmma_f32_16x16x4_f32(
        false, a, false, b0, (short)0, acc0, false, false);
    acc1 = __builtin_amdgcn_wmma_f32_16x16x4_f32(
        false, a, false, b1, (short)0, acc1, false, false);
  }

  // D layout: VGPR r -> row (r + 8*(lane>>4)), col (lane&15)
  const int hi = lane >> 4;
  const int n  = lane & 15;
  float* tout0 = t + (size_t)(tileM * 16) * RR + tnp * 32 + n;
  float* tout1 = tout0 + 16;
  #pragma unroll
  for (int r = 0; r < 8; ++r) {
    tout0[(size_t)(r + 8 * hi) * RR] = acc0[r];
    tout1[(size_t)(r + 8 * hi) * RR] = acc1[r];
  }
}

// ---------------------------------------------------------------------------
// Kernel 2: row norms  n[i] = sum_k t[i][k]^2   (4096 rows x 128)
// ---------------------------------------------------------------------------
__global__ __launch_bounds__(256) void norms_kernel(
    const float* __restrict__ t, float* __restrict__ norms)
{
  const int row = blockIdx.x * blockDim.x + threadIdx.x;  // 0..4095
  const float4* p = (const float4*)(t + (size_t)row * RR);
  float s = 0.f;
  #pragma unroll
  for (int i = 0; i < RR / 4; ++i) {
    float4 v = p[i];
    s += v.x * v.x + v.y * v.y + v.z * v.z + v.w * v.w;
  }
  norms[row] = s;
}

// ---------------------------------------------------------------------------
// Kernel 3: per batch b, G = t_b @ t_b^T ; out = n_i + n_j - 2G.
// One wave computes two adjacent 16x16 output tiles (same ti, tj pair):
// per K-step 1 A load + 2 B loads feed 2 WMMAs. K = 128 -> 64 WMMAs/wave.
// 16 batches * 16 ti * 8 tj-pairs = 2048 waves -> 256 blocks.
// ---------------------------------------------------------------------------
__global__ __launch_bounds__(256) void gram_wmma_kernel(
    const float* __restrict__ t,      // [4096, 128]
    const float* __restrict__ norms,  // [4096]
    float* __restrict__ out)          // [16, 256, 256]
{
  const int lane = threadIdx.x & 31;
  const int wid  = blockIdx.x * 8 + (threadIdx.x >> 5);  // 0..2047
  const int b    = wid >> 7;          // 0..15
  const int ti   = (wid >> 3) & 15;   // 0..15 row tile
  const int tjp  = wid & 7;           // 0..7  col tile pair

  const int m    = lane & 15;
  const int hi   = lane >> 4;
  const int koff = hi * 2;

  const float* tb    = t + (size_t)b * LL * RR;
  const float* arow  = tb + (size_t)(ti * 16 + m) * RR;        // A rows (i)
  const float* brow0 = tb + (size_t)(tjp * 32 + m) * RR;       // B for tile 2*tjp
  const float* brow1 = tb + (size_t)(tjp * 32 + 16 + m) * RR;  // B for tile 2*tjp+1

  v8f acc0 = {0.f, 0.f, 0.f, 0.f, 0.f, 0.f, 0.f, 0.f};
  v8f acc1 = {0.f, 0.f, 0.f, 0.f, 0.f, 0.f, 0.f, 0.f};

  #pragma unroll 8
  for (int k = 0; k < RR; k += 4) {
    v2f a  = *(const v2f*)(arow + k + koff);
    v2f b0 = *(const v2f*)(brow0 + k + koff);
    v2f b1 = *(const v2f*)(brow1 + k + koff);
    acc0 = __builtin_amdgcn_wmma_f32_16x16x4_f32(
        false, a, false, b0, (short)0, acc0, false, false);
    acc1 = __builtin_amdgcn_wmma_f32_16x16x4_f32(
        false, a, false, b1, (short)0, acc1, false, false);
  }

  // Epilogue: out[b][ti*16 + r + 8*hi][tj*16 + n] = n_i + n_j - 2*G
  const int n  = lane & 15;
  const float nj0 = norms[b * LL + tjp * 32 + n];
  const float nj1 = norms[b * LL + tjp * 32 + 16 + n];
  float* obase0 = out + (size_t)b * OUT_STRIDE + (size_t)(ti * 16) * LL + tjp * 32 + n;
  float* obase1 = obase0 + 16;
  #pragma unroll
  for (int r = 0; r < 8; ++r) {
    const int rr = r + 8 * hi;
    const float ni = norms[b * LL + ti * 16 + rr];
    obase0[(size_t)rr * LL] = ni + nj0 - 2.0f * acc0[r];
    obase1[(size_t)rr * LL] = ni + nj1 - 2.0f * acc1[r];
  }
}

// ---------------------------------------------------------------------------
extern "C" void kernel_launch(void* const* d_in, const int* in_sizes, int n_in,
                              void* d_out, int out_size, void* d_ws, size_t ws_size,
                              hipStream_t stream) {
  const float* batch = (const float*)d_in[0];  // [16, 256, 1024] f32
  const float* proj  = (const float*)d_in[1];  // [1024, 128] f32
  float* out = (float*)d_out;                  // [16, 256, 256] f32

  // Workspace layout: t [4096*128] f32 (2 MB) then norms [4096] f32 (16 KB)
  float* t     = (float*)d_ws;
  float* norms = t + (size_t)ROWS * RR;

  // Kernel 1: projection (1024 waves, 8 waves/block)
  proj_wmma_kernel<<<128, 256, 0, stream>>>(batch, proj, t);
  // Kernel 2: row norms (4096 rows)
  norms_kernel<<<ROWS / 256, 256, 0, stream>>>(t, norms);
  // Kernel 3: Gram + epilogue (2048 waves, 8 waves/block)
  gram_wmma_kernel<<<256, 256, 0, stream>>>(t, norms, out);
}